// GNN_30700426232195
// MI455X (gfx1250) — compile-verified
//
#include <hip/hip_runtime.h>

#define N_NODES 20000
#define N_EDGES 320000
#define D_IN    256
#define D_EDGE  64
#define D_HID   256
#define LN_EPS  1e-5f

typedef __attribute__((ext_vector_type(2))) float v2f;
typedef __attribute__((ext_vector_type(8))) float v8f;

// D = A(16x4 f32) * B(4x16 f32) + C(16x16 f32), full fp32 — matches reference numerics.
__device__ __forceinline__ v8f wmma_f32(v2f a, v2f b, v8f c) {
  return __builtin_amdgcn_wmma_f32_16x16x4_f32(false, a, false, b, (short)0, c, false, false);
}

__global__ void zero_kernel(float* __restrict__ p, int n) {
  int i = blockIdx.x * blockDim.x + threadIdx.x;
  if (i < n) p[i] = 0.0f;
}

__global__ void count_kernel(const int* __restrict__ dst, float* __restrict__ cnt, int ne) {
  int e = blockIdx.x * blockDim.x + threadIdx.x;
  if (e < ne) unsafeAtomicAdd(&cnt[dst[e]], 1.0f);
}

// Pack W[K x 256] row-pairs per column: Wp[p*256+n] = {W[2p][n], W[2p+1][n]}.
// Turns the two strided b32 B-fragment loads into one coalesced b64 load.
__global__ void pack_w_kernel(const float* __restrict__ W, v2f* __restrict__ Wp, int npairs) {
  int i = blockIdx.x * blockDim.x + threadIdx.x;
  if (i < npairs * D_HID) {
    int p = i >> 8;
    int n = i & 255;
    v2f v;
    v.x = W[(size_t)(2 * p) * D_HID + n];
    v.y = W[(size_t)(2 * p + 1) * D_HID + n];
    Wp[i] = v;
  }
}

// Y[M x 256] = X[M x K] @ W[K x 256] + bias.  One wave -> 16 rows x 64 cols (4 wmma tiles).
template <int K>
__global__ void gemm_bias_kernel(const float* __restrict__ X, const v2f* __restrict__ Wp,
                                 const float* __restrict__ bias, float* __restrict__ Y) {
  const int wave  = (blockIdx.x * blockDim.x + threadIdx.x) >> 5;
  const int lane  = threadIdx.x & 31;
  const int strip = wave >> 2;          // 16-row strip
  const int cg    = wave & 3;           // 64-col group
  const int half  = lane >> 4;
  const int mrow  = lane & 15;
  const int arow  = strip * 16 + mrow;  // A row this lane loads
  const int kb    = half * 2;           // K sub-offset per ISA f32 A/B layout
  const int nc0   = cg * 64 + mrow;     // base output column for this lane

  v8f acc0 = {}, acc1 = {}, acc2 = {}, acc3 = {};
#pragma unroll 4
  for (int k = 0; k < K; k += 4) {
    v2f a = *(const v2f*)(X + (size_t)arow * K + k + kb);
    const v2f* Wk = Wp + (size_t)((k >> 1) + half) * D_HID;
    v2f b0 = Wk[nc0];
    v2f b1 = Wk[nc0 + 16];
    v2f b2 = Wk[nc0 + 32];
    v2f b3 = Wk[nc0 + 48];
    acc0 = wmma_f32(a, b0, acc0);
    acc1 = wmma_f32(a, b1, acc1);
    acc2 = wmma_f32(a, b2, acc2);
    acc3 = wmma_f32(a, b3, acc3);
  }
  const float bv0 = bias[nc0], bv1 = bias[nc0 + 16], bv2 = bias[nc0 + 32], bv3 = bias[nc0 + 48];
#pragma unroll
  for (int r = 0; r < 8; r++) {
    int orow = strip * 16 + r + 8 * half;
    float* Yr = Y + (size_t)orow * D_HID + nc0;
    Yr[0]  = acc0[r] + bv0;
    Yr[16] = acc1[r] + bv1;
    Yr[32] = acc2[r] + bv2;
    Yr[48] = acc3[r] + bv3;
  }
}

// Fused edge pipeline: et = E @ We + be ; m = hn[src] + et ; atomic scatter-add m into S[dst].
// The 327MB message tensor never touches memory; atomics are 64B-coalesced (same edge row
// across each half-wave, 16 consecutive columns). hn and S are L2-resident (20.5MB each).
__global__ void edge_gemm_scatter_kernel(const float* __restrict__ E, const v2f* __restrict__ Wp,
                                         const float* __restrict__ be, const float* __restrict__ hn,
                                         const int* __restrict__ src, const int* __restrict__ dst,
                                         float* __restrict__ S) {
  const int wave  = (blockIdx.x * blockDim.x + threadIdx.x) >> 5;
  const int lane  = threadIdx.x & 31;
  const int strip = wave >> 2;
  const int cg    = wave & 3;
  const int half  = lane >> 4;
  const int mrow  = lane & 15;
  const int arow  = strip * 16 + mrow;
  const int kb    = half * 2;
  const int nc0   = cg * 64 + mrow;

  v8f acc0 = {}, acc1 = {}, acc2 = {}, acc3 = {};
#pragma unroll
  for (int k = 0; k < D_EDGE; k += 4) {
    v2f a = *(const v2f*)(E + (size_t)arow * D_EDGE + k + kb);
    const v2f* Wk = Wp + (size_t)((k >> 1) + half) * D_HID;
    v2f b0 = Wk[nc0];
    v2f b1 = Wk[nc0 + 16];
    v2f b2 = Wk[nc0 + 32];
    v2f b3 = Wk[nc0 + 48];
    acc0 = wmma_f32(a, b0, acc0);
    acc1 = wmma_f32(a, b1, acc1);
    acc2 = wmma_f32(a, b2, acc2);
    acc3 = wmma_f32(a, b3, acc3);
  }
  const float bv0 = be[nc0], bv1 = be[nc0 + 16], bv2 = be[nc0 + 32], bv3 = be[nc0 + 48];
#pragma unroll
  for (int r = 0; r < 8; r++) {
    int erow = strip * 16 + r + 8 * half;   // uniform across each half-wave
    int se = src[erow];
    int de = dst[erow];
    const float* hrow = hn + (size_t)se * D_HID + nc0;  // L2-resident gather
    float*       srow = S  + (size_t)de * D_HID + nc0;  // L2-resident scatter
    unsafeAtomicAdd(srow + 0,  acc0[r] + bv0 + hrow[0]);
    unsafeAtomicAdd(srow + 16, acc1[r] + bv1 + hrow[16]);
    unsafeAtomicAdd(srow + 32, acc2[r] + bv2 + hrow[32]);
    unsafeAtomicAdd(srow + 48, acc3[r] + bv3 + hrow[48]);
  }
}

// Z = concat(hn, s/max(cnt,1)) @ Wc + bc   (K = 512, the agg half scaled on the fly)
__global__ void concat_gemm_kernel(const float* __restrict__ hn, const float* __restrict__ S,
                                   const float* __restrict__ cnt, const v2f* __restrict__ Wp,
                                   const float* __restrict__ bc, float* __restrict__ Z) {
  const int wave  = (blockIdx.x * blockDim.x + threadIdx.x) >> 5;
  const int lane  = threadIdx.x & 31;
  const int strip = wave >> 2;
  const int cg    = wave & 3;
  const int half  = lane >> 4;
  const int mrow  = lane & 15;
  const int arow  = strip * 16 + mrow;
  const int kb    = half * 2;
  const int nc0   = cg * 64 + mrow;

  const float invc = 1.0f / fmaxf(cnt[arow], 1.0f);
  v8f acc0 = {}, acc1 = {}, acc2 = {}, acc3 = {};
#pragma unroll 4
  for (int k = 0; k < 2 * D_HID; k += 4) {
    v2f a;
    if (k < D_HID) {
      a = *(const v2f*)(hn + (size_t)arow * D_HID + k + kb);
    } else {
      a = *(const v2f*)(S + (size_t)arow * D_HID + (k - D_HID) + kb);
      a.x *= invc;
      a.y *= invc;
    }
    const v2f* Wk = Wp + (size_t)((k >> 1) + half) * D_HID;
    v2f b0 = Wk[nc0];
    v2f b1 = Wk[nc0 + 16];
    v2f b2 = Wk[nc0 + 32];
    v2f b3 = Wk[nc0 + 48];
    acc0 = wmma_f32(a, b0, acc0);
    acc1 = wmma_f32(a, b1, acc1);
    acc2 = wmma_f32(a, b2, acc2);
    acc3 = wmma_f32(a, b3, acc3);
  }
  const float bv0 = bc[nc0], bv1 = bc[nc0 + 16], bv2 = bc[nc0 + 32], bv3 = bc[nc0 + 48];
#pragma unroll
  for (int r = 0; r < 8; r++) {
    int orow = strip * 16 + r + 8 * half;
    float* Zr = Z + (size_t)orow * D_HID + nc0;
    Zr[0]  = acc0[r] + bv0;
    Zr[16] = acc1[r] + bv1;
    Zr[32] = acc2[r] + bv2;
    Zr[48] = acc3[r] + bv3;
  }
}

// In-place LayerNorm + ReLU, one 256-thread block per row.
__global__ void ln_relu_kernel(float* __restrict__ X, const float* __restrict__ g,
                               const float* __restrict__ b) {
  __shared__ float red[D_HID];
  const int row = blockIdx.x;
  const int c   = threadIdx.x;
  float x = X[(size_t)row * D_HID + c];
  red[c] = x;
  __syncthreads();
  for (int off = D_HID / 2; off > 0; off >>= 1) {
    if (c < off) red[c] += red[c + off];
    __syncthreads();
  }
  float mu = red[0] * (1.0f / D_HID);
  __syncthreads();
  float d = x - mu;
  red[c] = d * d;
  __syncthreads();
  for (int off = D_HID / 2; off > 0; off >>= 1) {
    if (c < off) red[c] += red[c + off];
    __syncthreads();
  }
  float var = red[0] * (1.0f / D_HID);
  float y = d * rsqrtf(var + LN_EPS) * g[c] + b[c];
  X[(size_t)row * D_HID + c] = fmaxf(y, 0.0f);
}

extern "C" void kernel_launch(void* const* d_in, const int* in_sizes, int n_in,
                              void* d_out, int out_size, void* d_ws, size_t ws_size,
                              hipStream_t stream) {
  const float* node_feat = (const float*)d_in[0];
  const float* edge_attr = (const float*)d_in[1];
  const int*   src       = (const int*)d_in[2];
  const int*   dst       = (const int*)d_in[3];
  const float* Wn        = (const float*)d_in[4];
  const float* bn        = (const float*)d_in[5];
  const float* We        = (const float*)d_in[6];
  const float* be        = (const float*)d_in[7];
  const float* Wc        = (const float*)d_in[8];
  const float* bc        = (const float*)d_in[9];
  const float* gamma     = (const float*)d_in[10];
  const float* beta      = (const float*)d_in[11];

  const size_t NH = (size_t)N_NODES * D_HID;
  float* ws   = (float*)d_ws;
  float* hn   = ws;            // N x H  node transform
  float* sacc = hn + NH;       // N x H  segment sum
  float* bufA = sacc + NH;     // N x H  layer output ping
  float* bufB = bufA + NH;     // N x H  layer output pong
  float* cnt  = bufB + NH;     // N      in-degree counts (f32)
  v2f*   WnP  = (v2f*)(cnt + N_NODES);                 // 3 * 128 * 256 pairs
  v2f*   WeP  = WnP + (size_t)3 * (D_IN / 2) * D_HID;  // 3 *  32 * 256 pairs
  v2f*   WcP  = WeP + (size_t)3 * (D_EDGE / 2) * D_HID;// 3 * 256 * 256 pairs

  // Pack all 3 layers' weights into row-pair-interleaved layout (pairs never
  // cross layer boundaries since per-layer K is even).
  {
    int pn = 3 * (D_IN / 2);    // 384 pairs
    int pe = 3 * (D_EDGE / 2);  // 96 pairs
    int pc = 3 * D_HID;         // 768 pairs
    pack_w_kernel<<<(pn * D_HID + 255) / 256, 256, 0, stream>>>(Wn, WnP, pn);
    pack_w_kernel<<<(pe * D_HID + 255) / 256, 256, 0, stream>>>(We, WeP, pe);
    pack_w_kernel<<<(pc * D_HID + 255) / 256, 256, 0, stream>>>(Wc, WcP, pc);
  }

  // Degree counts are layer-invariant; recomputed each call for determinism.
  zero_kernel<<<(N_NODES + 255) / 256, 256, 0, stream>>>(cnt, N_NODES);
  count_kernel<<<(N_EDGES + 255) / 256, 256, 0, stream>>>(dst, cnt, N_EDGES);

  const int node_blocks = ((N_NODES / 16) * 4) / 8;  // 625
  const int edge_blocks = ((N_EDGES / 16) * 4) / 8;  // 10000

  const float* h = node_feat;
  for (int l = 0; l < 3; l++) {
    float* hout = (l == 0) ? bufA : (l == 1) ? bufB : (float*)d_out;

    zero_kernel<<<((int)NH + 255) / 256, 256, 0, stream>>>(sacc, (int)NH);

    gemm_bias_kernel<D_IN><<<node_blocks, 256, 0, stream>>>(
        h, WnP + (size_t)l * (D_IN / 2) * D_HID, bn + l * D_HID, hn);

    edge_gemm_scatter_kernel<<<edge_blocks, 256, 0, stream>>>(
        edge_attr, WeP + (size_t)l * (D_EDGE / 2) * D_HID, be + l * D_HID, hn, src, dst, sacc);

    concat_gemm_kernel<<<node_blocks, 256, 0, stream>>>(
        hn, sacc, cnt, WcP + (size_t)l * D_HID * D_HID, bc + l * D_HID, hout);

    ln_relu_kernel<<<N_NODES, 256, 0, stream>>>(hout, gamma + l * D_HID, beta + l * D_HID);

    h = hout;
  }
}